// AULoss_71528385348329
// MI455X (gfx1250) — compile-verified
//
#include <hip/hip_runtime.h>

typedef float v2f __attribute__((ext_vector_type(2)));
typedef float v8f __attribute__((ext_vector_type(8)));

#define A_CLS   64
#define V_VIEW  128
#define D_DIM   128
#define N_ROWS  (A_CLS * V_VIEW)   // 8192
#define BT      128                // output tile (M = N)
#define KC      32                 // K chunk held in LDS
#define LDST    36                 // padded LDS row stride (floats): bank-conflict-free

// ff row i -> feats_[(i % 64)][i / 64][:]  (ff = transpose(feats_, (1,0,2)).reshape)
__device__ __forceinline__ const float* row_ptr(const float* feats, int i) {
    return feats + (size_t)(((i & 63) * V_VIEW) + (i >> 6)) * D_DIM;
}

// ---- kernel 1: zero accumulators, compute negative-pair count from labels ----
__global__ void au_init(const int* __restrict__ labels, float* __restrict__ ws) {
    if (threadIdx.x == 0) {
        int cnt[19];
        for (int c = 0; c < 19; ++c) cnt[c] = 0;
        for (int i = 0; i < A_CLS; ++i) {
            int l = labels[i];
            if (l >= 0 && l < 19) cnt[l]++;
        }
        float sum2 = 0.0f;
        for (int c = 0; c < 19; ++c) sum2 += (float)cnt[c] * (float)cnt[c];
        float n = (float)N_ROWS;
        ws[0] = 0.0f;                                            // align sum (full-sym)
        ws[1] = 0.0f;                                            // neg exp sum
        ws[2] = n * n - (float)V_VIEW * (float)V_VIEW * sum2;    // #negative pairs
    }
}

// ---- kernel 2: row norms n[i] = ||ff[i]||^2, one wave32 per row ----
__global__ __launch_bounds__(256) void au_norms(const float* __restrict__ feats,
                                                float* __restrict__ norms) {
    int wv   = threadIdx.x >> 5;
    int lane = threadIdx.x & 31;
    int row  = blockIdx.x * 8 + wv;
    const float4* p = (const float4*)row_ptr(feats, row);
    float4 v = p[lane];                      // 32 lanes x 4 = 128 floats
    float s = v.x * v.x + v.y * v.y + v.z * v.z + v.w * v.w;
    #pragma unroll
    for (int off = 16; off > 0; off >>= 1) s += __shfl_xor(s, off, 32);
    if (lane == 0) norms[row] = s;
}

// ---- kernel 3: fused WMMA Gram + AULoss epilogue over one 128x128 tile ----
__global__ __launch_bounds__(256) void au_gram(const float* __restrict__ feats,
                                               const float* __restrict__ norms,
                                               const int*  __restrict__ labels,
                                               float* __restrict__ ws) {
    __shared__ float lds_a[BT * LDST];
    __shared__ float lds_b[BT * LDST];
    __shared__ float nr_s[BT];
    __shared__ float nc_s[BT];
    __shared__ int   lab_s[A_CLS];
    __shared__ float red[16];

    const int t    = threadIdx.x;
    const int rb   = blockIdx.x >> 6;
    const int cb   = blockIdx.x & 63;
    const int r0   = rb * BT;
    const int c0   = cb * BT;
    const int lane = t & 31;
    const int wv   = t >> 5;             // 8 waves; wave w owns rows [16w,16w+16)
    const int mrow  = lane & 15;         // M (and N) sub-index held by this lane
    const int khalf = (lane >> 4) << 1;  // 0 or 2: K-pair selected by lane half

    if (t < A_CLS) lab_s[t] = labels[t];
    if (t < BT)    nr_s[t] = norms[r0 + t];
    else           nc_s[t - BT] = norms[c0 + (t - BT)];

    v8f acc[8];
    #pragma unroll
    for (int n = 0; n < 8; ++n) acc[n] = (v8f)0.0f;

    for (int kb = 0; kb < D_DIM; kb += KC) {
        __syncthreads();   // protect LDS reuse; also covers nr/nc/lab on iter 0
        // stage KC columns of both row-block and col-block into LDS
        for (int s = t; s < BT * (KC / 4); s += 256) {
            int row = s >> 3;
            int k4  = (s & 7) << 2;
            int o   = row * LDST + k4;
            float4 va = *(const float4*)(row_ptr(feats, r0 + row) + kb + k4);
            lds_a[o + 0] = va.x; lds_a[o + 1] = va.y;
            lds_a[o + 2] = va.z; lds_a[o + 3] = va.w;
            float4 vb = *(const float4*)(row_ptr(feats, c0 + row) + kb + k4);
            lds_b[o + 0] = vb.x; lds_b[o + 1] = vb.y;
            lds_b[o + 2] = vb.z; lds_b[o + 3] = vb.w;
        }
        __syncthreads();

        const float* pa = &lds_a[(wv * 16 + mrow) * LDST + khalf];
        #pragma unroll
        for (int kk = 0; kk < KC; kk += 4) {
            v2f av = *(const v2f*)(pa + kk);
            #pragma unroll
            for (int n = 0; n < 8; ++n) {
                v2f bv = *(const v2f*)&lds_b[(n * 16 + mrow) * LDST + kk + khalf];
                // D = A(16x4,f32) x B(4x16,f32) + C ; emits v_wmma_f32_16x16x4_f32
                acc[n] = __builtin_amdgcn_wmma_f32_16x16x4_f32(
                    false, av, false, bv, (short)0, acc[n], false, false);
            }
        }
    }

    // epilogue: d2 = max(n_r + n_c - 2G, 0); masked exp + align sums
    float negsum = 0.0f, alignsum = 0.0f;
    const int rhalf = (lane >> 4) << 3;  // lanes 16-31 hold rows M+8..M+15
    #pragma unroll
    for (int n = 0; n < 8; ++n) {
        int   cl = n * 16 + mrow;
        float nc = nc_s[cl];
        int   lc = lab_s[cl & 63];       // (c0+cl) % 64 == cl % 64 (c0 mult of 128)
        #pragma unroll
        for (int v = 0; v < 8; ++v) {
            int   rl = wv * 16 + rhalf + v;
            float d2 = fmaxf(nr_s[rl] + nc - 2.0f * acc[n][v], 0.0f);
            if (lab_s[rl & 63] != lc) negsum += __expf(-2.0f * d2);   // T = 2
            if ((rl & 63) == (cl & 63)) alignsum += d2;               // same-class pair
        }
    }
    #pragma unroll
    for (int off = 16; off > 0; off >>= 1) {
        negsum   += __shfl_xor(negsum, off, 32);
        alignsum += __shfl_xor(alignsum, off, 32);
    }
    __syncthreads();
    if (lane == 0) { red[wv] = negsum; red[8 + wv] = alignsum; }
    __syncthreads();
    if (t == 0) {
        float sn = 0.0f, sa = 0.0f;
        for (int i = 0; i < 8; ++i) { sn += red[i]; sa += red[8 + i]; }
        atomicAdd(&ws[1], sn);
        atomicAdd(&ws[0], sa);
    }
}

// ---- kernel 4: combine ----
__global__ void au_finish(const float* __restrict__ ws, float* __restrict__ out) {
    if (threadIdx.x == 0 && blockIdx.x == 0) {
        // full-sym align sum = 2 * triangular; mean over 64 classes of sum/8128
        float align   = ws[0] / (2.0f * 8128.0f * 64.0f);
        float uniform = __logf(ws[1] / ws[2]);
        out[0] = align + uniform;   // LOSS_WEIGHT = 1
    }
}

extern "C" void kernel_launch(void* const* d_in, const int* in_sizes, int n_in,
                              void* d_out, int out_size, void* d_ws, size_t ws_size,
                              hipStream_t stream) {
    const float* feats  = (const float*)d_in[0];
    const int*   labels = (const int*)d_in[1];
    float* ws    = (float*)d_ws;
    float* norms = ws + 16;   // 8192 floats

    au_init  <<<1, 64, 0, stream>>>(labels, ws);
    au_norms <<<N_ROWS / 8, 256, 0, stream>>>(feats, norms);
    au_gram  <<<64 * 64, 256, 0, stream>>>(feats, norms, labels, ws);
    au_finish<<<1, 1, 0, stream>>>(ws, (float*)d_out);
}